// Attention_43181601194684
// MI455X (gfx1250) — compile-verified
//
#include <hip/hip_runtime.h>

typedef float v2f __attribute__((ext_vector_type(2)));
typedef float v8f __attribute__((ext_vector_type(8)));

#define B_ 32
#define L_ 4096
#define H_ 512

// ---------------------------------------------------------------------------
// Phase 1: v[b,h] = sum_o h_last[b,o] * W[o,h]   (B=32 x K=512 x H=512)
// Uses V_WMMA_F32_16X16X4_F32. One wave per 16x16 output tile:
//   2 M-tiles x 32 N-tiles = 64 waves, 128 wmma ops each (K=512 / 4).
// Fragment layouts per CDNA5 ISA 7.12.2:
//   A 16x4:  vgpr0 = A[lane&15, K0 + 2*(lane>=16)], vgpr1 = next K
//   B 4x16:  vgpr0 = B[K0 + 2*(lane>=16), lane&15], vgpr1 = next K row
//   C 16x16: vgpr r = C[r + 8*(lane>=16), lane&15]
// ---------------------------------------------------------------------------
__global__ __launch_bounds__(32) void fold_hW_wmma(const float* __restrict__ hidden,
                                                   const float* __restrict__ W,
                                                   float* __restrict__ v) {
    const int lane = threadIdx.x;          // 0..31
    const int m    = lane & 15;
    const int hi   = lane >> 4;            // 0 or 1
    const int tile = blockIdx.x;           // 0..63
    const int mt   = tile >> 5;            // 0..1   (batch rows 0-15 / 16-31)
    const int nt   = tile & 31;            // 0..31  (H columns, 16 each)

    // hidden layout (2,2,B,H): h_last = hidden[0,1,:,:] -> offset B_*H_
    const float* __restrict__ hlast = hidden + (size_t)B_ * H_;
    const int row = mt * 16 + m;           // batch row this lane contributes to A
    const int col = nt * 16 + m;           // output column this lane handles for B/C/D

    v8f acc = {};
    for (int k = 0; k < H_; k += 4) {
        const int ka = k + 2 * hi;
        v2f a, b;
        a.x = hlast[row * H_ + ka];
        a.y = hlast[row * H_ + ka + 1];
        b.x = W[(size_t)ka * H_ + col];
        b.y = W[(size_t)(ka + 1) * H_ + col];
        acc = __builtin_amdgcn_wmma_f32_16x16x4_f32(
            /*neg_a=*/false, a, /*neg_b=*/false, b,
            /*c_mod=*/(short)0, acc, /*reuse_a=*/false, /*reuse_b=*/false);
    }

#pragma unroll
    for (int r = 0; r < 8; ++r) {
        const int orow = mt * 16 + r + 8 * hi;
        v[(size_t)orow * H_ + col] = acc[r];
    }
}

// ---------------------------------------------------------------------------
// Phase 2: energies[b,l] = v[b] . enc[b,l]   (pure bandwidth: 256 MB of enc)
// One wave per (b,l). Block = 8 waves sharing one batch row's v[] in LDS.
// Each lane reads 4 float4s of enc -> 32 lanes x 16 B = 512 B contiguous
// per iteration, fully coalesced. 5-step shuffle reduction (wave32).
// ---------------------------------------------------------------------------
__global__ __launch_bounds__(256) void energies_kernel(const float* __restrict__ enc,
                                                       const float* __restrict__ v,
                                                       float* __restrict__ energ) {
    __shared__ float vsh[H_];
    const int b   = blockIdx.x >> 9;       // L_/8 = 512 l-tiles per batch
    const int lt  = blockIdx.x & 511;
    const int tid = threadIdx.x;

    // stage v[b][0..511] into LDS (256 threads x float2)
    ((float2*)vsh)[tid] = ((const float2*)(v + (size_t)b * H_))[tid];
    __syncthreads();

    const int wave = tid >> 5;
    const int lane = tid & 31;
    const int l    = lt * 8 + wave;

    const float4* __restrict__ e  = (const float4*)(enc + ((size_t)b * L_ + l) * H_);
    const float4* __restrict__ vs = (const float4*)vsh;

    float acc = 0.f;
#pragma unroll
    for (int k = 0; k < 4; ++k) {
        const int idx = k * 32 + lane;     // 128 float4s per row
        const float4 ev = e[idx];
        const float4 vv = vs[idx];
        acc += ev.x * vv.x + ev.y * vv.y + ev.z * vv.z + ev.w * vv.w;
    }
#pragma unroll
    for (int m = 16; m > 0; m >>= 1) acc += __shfl_xor(acc, m, 32);

    if (lane == 0) energ[(size_t)b * L_ + l] = acc;
}

// ---------------------------------------------------------------------------
// Phase 3: row-wise softmax over L=4096, in place on d_out (B rows).
// 256 threads x 16 values each, kept in registers; LDS tree reductions.
// (h.bias term omitted: constant per row, cancels exactly in softmax.)
// ---------------------------------------------------------------------------
__global__ __launch_bounds__(256) void softmax_kernel(float* __restrict__ out) {
    __shared__ float red[256];
    const int b   = blockIdx.x;
    const int tid = threadIdx.x;
    float* __restrict__ row = out + (size_t)b * L_;

    float vals[16];
    float mx = -3.402823466e38f;
#pragma unroll
    for (int i = 0; i < 16; ++i) {
        vals[i] = row[tid + i * 256];
        mx = fmaxf(mx, vals[i]);
    }
    red[tid] = mx;
    __syncthreads();
    for (int s = 128; s > 0; s >>= 1) {
        if (tid < s) red[tid] = fmaxf(red[tid], red[tid + s]);
        __syncthreads();
    }
    mx = red[0];
    __syncthreads();

    float sum = 0.f;
#pragma unroll
    for (int i = 0; i < 16; ++i) {
        vals[i] = __expf(vals[i] - mx);
        sum += vals[i];
    }
    red[tid] = sum;
    __syncthreads();
    for (int s = 128; s > 0; s >>= 1) {
        if (tid < s) red[tid] += red[tid + s];
        __syncthreads();
    }
    const float inv = 1.f / red[0];
#pragma unroll
    for (int i = 0; i < 16; ++i) row[tid + i * 256] = vals[i] * inv;
}

// ---------------------------------------------------------------------------
extern "C" void kernel_launch(void* const* d_in, const int* in_sizes, int n_in,
                              void* d_out, int out_size, void* d_ws, size_t ws_size,
                              hipStream_t stream) {
    const float* hidden = (const float*)d_in[0];   // (2,2,B,H)
    const float* enc    = (const float*)d_in[1];   // (B,L,H)
    const float* W      = (const float*)d_in[2];   // (H,H)
    // d_in[3] = bias: softmax-invariant, intentionally unused.
    float* out = (float*)d_out;                    // (B,1,L) = B*L floats
    float* v   = (float*)d_ws;                     // B*H floats = 64 KB scratch

    fold_hW_wmma<<<64, 32, 0, stream>>>(hidden, W, v);
    energies_kernel<<<B_ * (L_ / 8), 256, 0, stream>>>(enc, v, out);
    softmax_kernel<<<B_, 256, 0, stream>>>(out);
}